// LearnableDCTNet_85169201479729
// MI455X (gfx1250) — compile-verified
//
#include <hip/hip_runtime.h>
#include <hip/hip_bf16.h>
#include <math.h>

typedef __attribute__((ext_vector_type(2))) float v2f;
typedef __attribute__((ext_vector_type(8))) float v8f;

#define PI_T      3.1415926f
#define NQ        16
#define NPIX      33554432      // 32*256*64*64
#define NTILES    131072        // 8192 images * 4*4 tiles of 16x16
#define WAVES_PB  8

// ---------------------------------------------------------------------------
// Phase 0: noise_level MLP (exact-erf GELU) + softmax -> learned_qmat (8x8)
// ---------------------------------------------------------------------------
__global__ void lq_kernel(const float* __restrict__ noise,
                          const float* __restrict__ qmat,
                          const float* __restrict__ qw_in,
                          const float* __restrict__ w1, const float* __restrict__ b1,
                          const float* __restrict__ w2, const float* __restrict__ b2,
                          const float* __restrict__ w3, const float* __restrict__ b3,
                          float* __restrict__ lq_out)
{
    __shared__ float coef[NQ];
    const int t = threadIdx.x;
    if (t == 0) {
        float nl = noise[0];
        float h1[16], h2[16], nr[18];
        for (int j = 0; j < 16; ++j) {
            float z = nl * w1[j] + b1[j];
            h1[j] = 0.5f * z * (1.0f + erff(z * 0.70710678118f));
        }
        for (int j = 0; j < 16; ++j) {
            float z = b2[j];
            for (int i = 0; i < 16; ++i) z += h1[i] * w2[i * 16 + j];
            h2[j] = 0.5f * z * (1.0f + erff(z * 0.70710678118f));
        }
        for (int j = 0; j < 18; ++j) {
            float z = b3[j];
            for (int i = 0; i < 16; ++i) z += h2[i] * w3[i * 18 + j];
            nr[j] = z;
        }
        float mx = qw_in[0];
        for (int i = 1; i < 16; ++i) mx = fmaxf(mx, qw_in[i]);
        float e[16], s = 0.0f;
        for (int i = 0; i < 16; ++i) { e[i] = expf(qw_in[i] - mx); s += e[i]; }
        for (int i = 0; i < 16; ++i)
            coef[i] = (e[i] / s * nr[0] + nr[1]) * nr[2 + i];
    }
    __syncthreads();
    if (t < 64) {
        float acc = 0.0f;
        for (int i = 0; i < 16; ++i) acc += coef[i] * qmat[i * 64 + t];
        lq_out[t] = acc;
    }
}

// ---------------------------------------------------------------------------
// Phase 1: per-wave 16x16 tile = 2x2 DCT blocks, 6 chained 16x16x16 matmuls
// via V_WMMA_F32_16X16X4_F32 (4 accumulating WMMAs each).
// ---------------------------------------------------------------------------
__global__ __launch_bounds__(256) void dct_kernel(const float* __restrict__ x,
                                                  const float* __restrict__ lq_in,
                                                  float* __restrict__ out_low,
                                                  float* __restrict__ out_high,
                                                  int nTiles)
{
    __shared__ float Cm[64];                 // 8x8 DCT basis
    __shared__ float lq[64];                 // learned quant matrix
    __shared__ float buf[WAVES_PB][3 * 256]; // wave-private staging (3x 16x16 f32)

    const int t = threadIdx.x;
    if (t < 64) {
        int i = t >> 3, j = t & 7;
        Cm[t] = (i == 0) ? 0.35355339059f
                         : cosf(PI_T * (float)(i * (2 * j + 1)) * (1.0f / 16.0f)) * 0.5f;
        lq[t] = lq_in[t];
    }
    __syncthreads();

    const int wave = t >> 5;
    const int lane = t & 31;
    const int half = lane >> 4;   // 0: lanes 0-15, 1: lanes 16-31
    const int l16  = lane & 15;
    const int col8 = lane & 7;

    const int tile = blockIdx.x * WAVES_PB + wave;
    if (tile >= nTiles) return;   // wave-uniform; grid is exact anyway

    // Chat = blockdiag(C, C): Chat(m,k)
    auto Chat = [&](int m, int k) -> float {
        return (((m ^ k) & 8) != 0) ? 0.0f : Cm[(m & 7) * 8 + (k & 7)];
    };

    // P = A-layout(Chat) == B-layout(Chat^T);  Q = A-layout(Chat^T) == B-layout(Chat)
    v2f P[4], Q[4];
    for (int kb = 0; kb < 4; ++kb) {
        int k0 = 4 * kb + 2 * half;
        P[kb].x = Chat(l16, k0);
        P[kb].y = Chat(l16, k0 + 1);
        Q[kb].x = Chat(k0,     l16);
        Q[kb].y = Chat(k0 + 1, l16);
    }

    // mask column for this lane's C/D-layout column (n % 8)
    float mcol[8];
    for (int v = 0; v < 8; ++v) mcol[v] = lq[v * 8 + col8];

    const int img  = tile >> 4;
    const int ti   = tile & 15;
    const int base = img * 4096 + (ti >> 2) * (16 * 64) + (ti & 3) * 16;

    float* b0 = &buf[wave][0];
    float* b1 = &buf[wave][256];
    float* b2 = &buf[wave][512];

    // ---- load X tile in B layout ------------------------------------------
    v2f XB[4];
    for (int kb = 0; kb < 4; ++kb) {
        int r = 4 * kb + 2 * half;
        XB[kb].x = x[base + r * 64 + l16];
        XB[kb].y = x[base + (r + 1) * 64 + l16];
    }

    // ---- S1 = Chat @ X ----------------------------------------------------
    v8f acc = {};
    for (int kb = 0; kb < 4; ++kb)
        acc = __builtin_amdgcn_wmma_f32_16x16x4_f32(false, P[kb], false, XB[kb],
                                                    (short)0, acc, false, false);
    for (int v = 0; v < 8; ++v) b0[(v + 8 * half) * 16 + l16] = acc[v];

    // ---- Y = S1 @ Chat^T  (A = S1 from LDS, B = P) ------------------------
    v2f AS[4];
    for (int kb = 0; kb < 4; ++kb) {
        int k0 = 4 * kb + 2 * half;
        AS[kb].x = b0[l16 * 16 + k0];
        AS[kb].y = b0[l16 * 16 + k0 + 1];
    }
    v8f Y = {};
    for (int kb = 0; kb < 4; ++kb)
        Y = __builtin_amdgcn_wmma_f32_16x16x4_f32(false, AS[kb], false, P[kb],
                                                  (short)0, Y, false, false);

    // ---- apply mask: Ylow -> b0, Yhigh -> b1 ------------------------------
    for (int v = 0; v < 8; ++v) {
        float ym = Y[v] * mcol[v];
        float yh = Y[v] * (1.0f - mcol[v]);
        b0[(v + 8 * half) * 16 + l16] = ym;
        b1[(v + 8 * half) * 16 + l16] = yh;
    }

    // ---- S3 = Chat^T @ Ylow  (A = Q, B from LDS) --------------------------
    v2f BB[4];
    for (int kb = 0; kb < 4; ++kb) {
        int k0 = 4 * kb + 2 * half;
        BB[kb].x = b0[k0 * 16 + l16];
        BB[kb].y = b0[(k0 + 1) * 16 + l16];
    }
    v8f S3 = {};
    for (int kb = 0; kb < 4; ++kb)
        S3 = __builtin_amdgcn_wmma_f32_16x16x4_f32(false, Q[kb], false, BB[kb],
                                                   (short)0, S3, false, false);
    for (int v = 0; v < 8; ++v) b2[(v + 8 * half) * 16 + l16] = S3[v];

    // ---- low = S3 @ Chat  (A = S3 from LDS, B = Q) ------------------------
    for (int kb = 0; kb < 4; ++kb) {
        int k0 = 4 * kb + 2 * half;
        AS[kb].x = b2[l16 * 16 + k0];
        AS[kb].y = b2[l16 * 16 + k0 + 1];
    }
    v8f LO = {};
    for (int kb = 0; kb < 4; ++kb)
        LO = __builtin_amdgcn_wmma_f32_16x16x4_f32(false, AS[kb], false, Q[kb],
                                                   (short)0, LO, false, false);
    for (int v = 0; v < 8; ++v)
        out_low[base + (v + 8 * half) * 64 + l16] = LO[v];

    // ---- S5 = Chat^T @ Yhigh ----------------------------------------------
    for (int kb = 0; kb < 4; ++kb) {
        int k0 = 4 * kb + 2 * half;
        BB[kb].x = b1[k0 * 16 + l16];
        BB[kb].y = b1[(k0 + 1) * 16 + l16];
    }
    v8f S5 = {};
    for (int kb = 0; kb < 4; ++kb)
        S5 = __builtin_amdgcn_wmma_f32_16x16x4_f32(false, Q[kb], false, BB[kb],
                                                   (short)0, S5, false, false);
    for (int v = 0; v < 8; ++v) b2[(v + 8 * half) * 16 + l16] = S5[v];

    // ---- high = S5 @ Chat --------------------------------------------------
    for (int kb = 0; kb < 4; ++kb) {
        int k0 = 4 * kb + 2 * half;
        AS[kb].x = b2[l16 * 16 + k0];
        AS[kb].y = b2[l16 * 16 + k0 + 1];
    }
    v8f HI = {};
    for (int kb = 0; kb < 4; ++kb)
        HI = __builtin_amdgcn_wmma_f32_16x16x4_f32(false, AS[kb], false, Q[kb],
                                                   (short)0, HI, false, false);
    for (int v = 0; v < 8; ++v)
        out_high[base + (v + 8 * half) * 64 + l16] = HI[v];
}

// ---------------------------------------------------------------------------
extern "C" void kernel_launch(void* const* d_in, const int* in_sizes, int n_in,
                              void* d_out, int out_size, void* d_ws, size_t ws_size,
                              hipStream_t stream) {
    const float* x     = (const float*)d_in[0];
    const float* noise = (const float*)d_in[1];
    const float* qmat  = (const float*)d_in[2];
    const float* qw    = (const float*)d_in[3];
    const float* w1    = (const float*)d_in[4];
    const float* b1    = (const float*)d_in[5];
    const float* w2    = (const float*)d_in[6];
    const float* b2    = (const float*)d_in[7];
    const float* w3    = (const float*)d_in[8];
    const float* b3    = (const float*)d_in[9];

    float* lq       = (float*)d_ws;          // 64 floats
    float* out_low  = (float*)d_out;
    float* out_high = out_low + NPIX;

    lq_kernel<<<1, 64, 0, stream>>>(noise, qmat, qw, w1, b1, w2, b2, w3, b3, lq);
    dct_kernel<<<NTILES / WAVES_PB, 256, 0, stream>>>(x, lq, out_low, out_high, NTILES);
}